// CombinedLoss_88476326298381
// MI455X (gfx1250) — compile-verified
//
#include <hip/hip_runtime.h>
#include <hip/hip_bf16.h>

// ---------------------------------------------------------------------------
// CombinedLoss (InfoNCE + distill KL) for MI455X / gfx1250, wave32, WMMA fp32
//   B=1024 queries, K=16 docs/query, D=1024, TEMP=0.02, ALPHA=0.4
//
// Kernel 1: each block = 8 waves, covers 32 query rows x 1024 doc cols.
//   Each wave owns a 32x32 macro-tile per pass: 2 M-tiles x 2 N-tiles
//   -> 4 fp32 WMMA accumulators, 2 A-loads + 2 B-loads per 4 WMMAs.
//   Q tile staged in LDS via GLOBAL_LOAD_ASYNC_TO_LDS_B128 (ASYNCcnt).
// ---------------------------------------------------------------------------

typedef __attribute__((ext_vector_type(2))) float v2f;
typedef __attribute__((ext_vector_type(8))) float v8f;

#define BQ      1024
#define KDOCS   16
#define DDIM    1024
#define NDOCS   (BQ * KDOCS)       // 16384
#define MROWS   32                 // query rows per block
#define QTILES  (BQ / MROWS)       // 32
#define NTILES  (NDOCS / 1024)     // 16 column blocks of 1024 docs
#define QSTRIDE 1028               // padded LDS row stride (floats) -> conflict-free
#define INV_TEMP 50.0f
#define ALPHA_C  0.4f

__device__ __forceinline__ void lse_merge(float& M, float& S, float m2, float s2) {
    float mn = fmaxf(M, m2);
    S = S * __expf(M - mn) + s2 * __expf(m2 - mn);
    M = mn;
}

// ---------------------------------------------------------------------------
__global__ void __launch_bounds__(256)
gemm_lse_partial(const float* __restrict__ q,
                 const float* __restrict__ docs,
                 float* __restrict__ wsM,
                 float* __restrict__ wsS) {
    extern __shared__ float smem[];
    float* ldsQ = smem;                       // MROWS * QSTRIDE floats
    float* ldsM = smem + MROWS * QSTRIDE;     // 8 waves * 32 rows
    float* ldsS = ldsM + 8 * MROWS;

    const int qt   = blockIdx.x;              // query tile (32 rows)
    const int nt   = blockIdx.y;              // doc column tile (1024 cols)
    const int tid  = threadIdx.x;
    const int wave = tid >> 5;
    const int lane = tid & 31;
    const int l16  = lane & 15;
    const int koff = (lane >> 4) << 1;        // 0 for lanes 0-15, 2 for lanes 16-31

    // ---- stage Q tile (32 x 1024 fp32) into LDS via async memory->LDS ----
    const float* qbase = q + (size_t)qt * MROWS * DDIM;
    for (int idx = tid; idx < MROWS * (DDIM / 4); idx += 256) {
        int r  = idx >> 8;                    // DDIM/4 == 256
        int c4 = idx & 255;
        unsigned int       laddr = (unsigned int)(uintptr_t)&ldsQ[r * QSTRIDE + (c4 << 2)];
        unsigned long long gaddr = (unsigned long long)(uintptr_t)(qbase + (size_t)r * DDIM + (c4 << 2));
        asm volatile("global_load_async_to_lds_b128 %0, %1, off"
                     :: "v"(laddr), "v"(gaddr) : "memory");
    }
    asm volatile("s_wait_asynccnt 0" ::: "memory");
    __syncthreads();

    // per-wave running (max, sum-exp): [mt][slot]; slot i -> row mt*16 + i (+8 for hi half)
    float mr[2][8], sr[2][8];
#pragma unroll
    for (int t = 0; t < 2; ++t)
#pragma unroll
        for (int i = 0; i < 8; ++i) { mr[t][i] = -__builtin_inff(); sr[t][i] = 0.0f; }

    const int ntBase = nt * 1024;
    const int aoff0 = l16 * QSTRIDE + koff;
    const int aoff1 = (l16 + 16) * QSTRIDE + koff;

    for (int pass = 0; pass < 4; ++pass) {
        const int col0 = ntBase + pass * 256 + wave * 32;
        const float* dptr0 = docs + (size_t)(col0 + l16) * DDIM + koff;        // N-tile 0
        const float* dptr1 = docs + (size_t)(col0 + 16 + l16) * DDIM + koff;   // N-tile 1
        if (pass < 3) {  // stream prefetch of next pass's doc rows
            __builtin_prefetch(docs + (size_t)(col0 + 256 + l16) * DDIM, 0, 1);
        }

        v8f acc00 = {}, acc01 = {}, acc10 = {}, acc11 = {};
#pragma unroll 4
        for (int k0 = 0; k0 < DDIM; k0 += 4) {
            v2f a0 = *(const v2f*)&ldsQ[aoff0 + k0];   // rows 0..15
            v2f a1 = *(const v2f*)&ldsQ[aoff1 + k0];   // rows 16..31
            v2f b0 = *(const v2f*)&dptr0[k0];
            v2f b1 = *(const v2f*)&dptr1[k0];
            acc00 = __builtin_amdgcn_wmma_f32_16x16x4_f32(false, a0, false, b0, (short)0, acc00, false, false);
            acc01 = __builtin_amdgcn_wmma_f32_16x16x4_f32(false, a0, false, b1, (short)0, acc01, false, false);
            acc10 = __builtin_amdgcn_wmma_f32_16x16x4_f32(false, a1, false, b0, (short)0, acc10, false, false);
            acc11 = __builtin_amdgcn_wmma_f32_16x16x4_f32(false, a1, false, b1, (short)0, acc11, false, false);
        }

        // online logsumexp update; combine the two N-tiles before lane-reduce
#pragma unroll
        for (int t = 0; t < 2; ++t) {
#pragma unroll
            for (int i = 0; i < 8; ++i) {
                float v0 = (t == 0 ? acc00[i] : acc10[i]) * INV_TEMP;
                float v1 = (t == 0 ? acc01[i] : acc11[i]) * INV_TEMP;
                float m = fmaxf(v0, v1);
                m = fmaxf(m, __shfl_xor(m, 1, 16));
                m = fmaxf(m, __shfl_xor(m, 2, 16));
                m = fmaxf(m, __shfl_xor(m, 4, 16));
                m = fmaxf(m, __shfl_xor(m, 8, 16));
                float e = __expf(v0 - m) + __expf(v1 - m);
                e += __shfl_xor(e, 1, 16);
                e += __shfl_xor(e, 2, 16);
                e += __shfl_xor(e, 4, 16);
                e += __shfl_xor(e, 8, 16);
                lse_merge(mr[t][i], sr[t][i], m, e);
            }
        }
    }

    // lanes 0 / 16 publish rows {mt*16 + 0..7} / {mt*16 + 8..15} for this wave
    if (l16 == 0) {
        int rbase = (lane >> 4) * 8;
#pragma unroll
        for (int t = 0; t < 2; ++t)
#pragma unroll
            for (int i = 0; i < 8; ++i) {
                ldsM[wave * MROWS + t * 16 + rbase + i] = mr[t][i];
                ldsS[wave * MROWS + t * 16 + rbase + i] = sr[t][i];
            }
    }
    __syncthreads();

    // combine 8 waves -> per-(qt, nt, row) partial (deterministic order)
    if (tid < MROWS) {
        float M = -__builtin_inff(), S = 0.0f;
        for (int w = 0; w < 8; ++w)
            lse_merge(M, S, ldsM[w * MROWS + tid], ldsS[w * MROWS + tid]);
        size_t o = ((size_t)qt * NTILES + nt) * MROWS + tid;
        wsM[o] = M;
        wsS[o] = S;
    }
}

// ---------------------------------------------------------------------------
// Kernel 2: in-group logits sim[b][k] = dot(Q[b], doc[b*K+k]) / TEMP
// ---------------------------------------------------------------------------
__global__ void __launch_bounds__(256)
group_sims(const float* __restrict__ q,
           const float* __restrict__ docs,
           float* __restrict__ wsSim) {
    const int b    = blockIdx.x;
    const int wave = threadIdx.x >> 5;
    const int lane = threadIdx.x & 31;
    const float4* qrow = (const float4*)(q + (size_t)b * DDIM);

    for (int k = wave; k < KDOCS; k += 8) {
        const float4* drow =
            (const float4*)(docs + ((size_t)b * KDOCS + k) * DDIM);
        float acc = 0.0f;
        for (int j = lane; j < DDIM / 4; j += 32) {
            float4 a = qrow[j];
            float4 d = drow[j];
            acc = fmaf(a.x, d.x, acc);
            acc = fmaf(a.y, d.y, acc);
            acc = fmaf(a.z, d.z, acc);
            acc = fmaf(a.w, d.w, acc);
        }
#pragma unroll
        for (int off = 16; off >= 1; off >>= 1)
            acc += __shfl_xor(acc, off, 32);
        if (lane == 0)
            wsSim[(size_t)b * KDOCS + k] = acc * INV_TEMP;
    }
}

// ---------------------------------------------------------------------------
// Kernel 3: combine partials -> lse per row; InfoNCE + distill KL; reduce.
// ---------------------------------------------------------------------------
__global__ void __launch_bounds__(1024)
finalize_loss(const float* __restrict__ wsM,
              const float* __restrict__ wsS,
              const float* __restrict__ wsSim,
              const float* __restrict__ soft_labels,
              float* __restrict__ out) {
    __shared__ float redI[32];
    __shared__ float redK[32];

    const int b    = threadIdx.x;       // 0..1023
    const int qt   = b >> 5;            // 32 rows per tile
    const int row  = b & 31;
    const int wave = threadIdx.x >> 5;
    const int lane = threadIdx.x & 31;

    // merge the 16 column-tile partials (fixed order -> deterministic)
    float M = -__builtin_inff(), S = 0.0f;
    for (int nt = 0; nt < NTILES; ++nt) {
        size_t o = ((size_t)qt * NTILES + nt) * MROWS + row;
        lse_merge(M, S, wsM[o], wsS[o]);
    }
    float lse  = __logf(S) + M;
    float pos  = wsSim[(size_t)b * KDOCS];   // k=0 is the positive
    float linf = lse - pos;

    // distill: KL(p_teacher || softmax(sim[b,:]))
    float smax = -__builtin_inff();
#pragma unroll
    for (int k = 0; k < KDOCS; ++k)
        smax = fmaxf(smax, wsSim[(size_t)b * KDOCS + k]);
    float z = 0.0f;
#pragma unroll
    for (int k = 0; k < KDOCS; ++k)
        z += __expf(wsSim[(size_t)b * KDOCS + k] - smax);
    float logZ = __logf(z) + smax;

    float tsum = 0.0f;
#pragma unroll
    for (int k = 0; k < KDOCS; ++k)
        tsum += soft_labels[(size_t)b * KDOCS + k];
    float inv = 1.0f / (tsum + 1e-9f);

    float kl = 0.0f;
#pragma unroll
    for (int k = 0; k < KDOCS; ++k) {
        float p     = soft_labels[(size_t)b * KDOCS + k] * inv;
        float logps = wsSim[(size_t)b * KDOCS + k] - logZ;
        float xlogy = (p > 0.0f) ? p * __logf(p) : 0.0f;   // xlogy semantics
        kl += xlogy - p * logps;
    }

    // two-level deterministic reduction over the 1024 rows
    float vi = linf, vk = kl;
#pragma unroll
    for (int off = 16; off >= 1; off >>= 1) {
        vi += __shfl_xor(vi, off, 32);
        vk += __shfl_xor(vk, off, 32);
    }
    if (lane == 0) { redI[wave] = vi; redK[wave] = vk; }
    __syncthreads();

    if (threadIdx.x < 32) {
        float a = redI[threadIdx.x];
        float c = redK[threadIdx.x];
#pragma unroll
        for (int off = 16; off >= 1; off >>= 1) {
            a += __shfl_xor(a, off, 32);
            c += __shfl_xor(c, off, 32);
        }
        if (threadIdx.x == 0) {
            float li = a / (float)BQ;     // mean InfoNCE
            float ld = c / (float)BQ;     // kl / B
            out[0] = (1.0f - ALPHA_C) * li + ALPHA_C * ld;   // total
            out[1] = li;                                     // loss_infonce
            out[2] = ld;                                     // loss_distill
        }
    }
}

// ---------------------------------------------------------------------------
extern "C" void kernel_launch(void* const* d_in, const int* in_sizes, int n_in,
                              void* d_out, int out_size, void* d_ws, size_t ws_size,
                              hipStream_t stream) {
    const float* q      = (const float*)d_in[0];   // [B, D]
    const float* docs   = (const float*)d_in[1];   // [B*K, D]
    const float* labels = (const float*)d_in[2];   // [B, K]
    // d_in[3] = num_docs_per_sample (uniform K) -- unused
    float* out = (float*)d_out;

    // workspace: (m,s) partials [32][16][32] + group sims [B*K]  = 192 KB
    float* wsM   = (float*)d_ws;
    float* wsS   = wsM + (size_t)QTILES * NTILES * MROWS;   // +16384
    float* wsSim = wsS + (size_t)QTILES * NTILES * MROWS;   // +16384

    size_t shmem = (size_t)(MROWS * QSTRIDE + 2 * 8 * MROWS) * sizeof(float);  // ~133.6 KB

    gemm_lse_partial<<<dim3(QTILES, NTILES), 256, shmem, stream>>>(
        q, docs, wsM, wsS);
    group_sims<<<BQ, 256, 0, stream>>>(q, docs, wsSim);
    finalize_loss<<<1, 1024, 0, stream>>>(wsM, wsS, wsSim, labels, out);
}